// MultiDimensionalGCN_27565100105911
// MI455X (gfx1250) — compile-verified
//
#include <hip/hip_runtime.h>

// ---------------------------------------------------------------------------
// MultiDimensionalGCN for MI455X (gfx1250, wave32, WMMA).
// All GEMMs use V_WMMA_F32_16X16X4_F32 (full f32 precision: the spectral
// eigengap threshold makes low-precision matrix math unsafe). Each wave
// register-blocks a 32x32 C tile (4 accumulators -> 4 independent WMMAs per
// K-step, 2x the arithmetic intensity of a single-tile wave). Scatter/degree
// use f32 global atomics; reductions use LDS; eigenvalues via Householder
// tridiagonalization + Sturm bisection (one workgroup per dim-matrix).
// ---------------------------------------------------------------------------

#define Dd  5
#define Nn  1024
#define Ee  65536
#define INF 768
#define Hh  512
#define Oo  256

typedef __attribute__((ext_vector_type(2))) float v2f;
typedef __attribute__((ext_vector_type(8))) float v8f;

static inline int cdiv(long a, long b) { return (int)((a + b - 1) / b); }

// ------------------------------ WMMA GEMM ----------------------------------
// C[d] = A[d] @ B[d] (+ bias[col]), batched over blockIdx.z = d.
// Block = 128 threads = 4 waves in a 2x2 arrangement -> 64x64 C tile.
// Wave computes a 32x32 tile: 2 A-frags x 2 B-frags -> 4 v_wmma per K-step
// on independent accumulators (no WMMA->WMMA RAW hazards). All M,Ncols,K are
// multiples of 64/4, so EXEC is all-ones everywhere (WMMA requirement).
__global__ __launch_bounds__(128) void k_gemm(
    const float* __restrict__ A, int lda, long sA,
    const float* __restrict__ B, int ldb, long sB,
    float* __restrict__ C, int ldc, long sC,
    const float* __restrict__ bias, int K, int transB)
{
  const int d = blockIdx.z;
  const float* Ad = A + (long)d * sA;
  const float* Bd = B + (long)d * sB;
  float*       Cd = C + (long)d * sC;

  const int wave = threadIdx.x >> 5;
  const int lane = threadIdx.x & 31;
  const int l16  = lane & 15;
  const int kh   = (lane >> 4) << 1;              // 0 (lanes 0-15) or 2 (lanes 16-31)
  const int m0   = (blockIdx.y << 6) + ((wave & 1) << 5);
  const int n0   = (blockIdx.x << 6) + ((wave >> 1) << 5);

  const float* arow0 = Ad + (long)(m0 + l16) * lda;        // M sub-tile 0
  const float* arow1 = arow0 + (long)16 * lda;             // M sub-tile 1
  v8f acc00 = {0.f,0.f,0.f,0.f,0.f,0.f,0.f,0.f};
  v8f acc01 = acc00, acc10 = acc00, acc11 = acc00;

  if (transB) {
    // S = A @ A^T style: B stored row-major [Ncols, K]
    const float* brow0 = Bd + (long)(n0 + l16) * ldb;
    const float* brow1 = brow0 + (long)16 * ldb;
    for (int k = 0; k < K; k += 4) {
      if ((k & 63) == 0) {
        __builtin_prefetch(arow0 + k + 128, 0, 3);   // global_prefetch_b8 (near)
        __builtin_prefetch(brow0 + k + 128, 0, 3);
      }
      v2f a0; a0.x = arow0[k + kh]; a0.y = arow0[k + kh + 1];
      v2f a1; a1.x = arow1[k + kh]; a1.y = arow1[k + kh + 1];
      v2f b0; b0.x = brow0[k + kh]; b0.y = brow0[k + kh + 1];
      v2f b1; b1.x = brow1[k + kh]; b1.y = brow1[k + kh + 1];
      acc00 = __builtin_amdgcn_wmma_f32_16x16x4_f32(false, a0, false, b0, (short)0, acc00, false, false);
      acc01 = __builtin_amdgcn_wmma_f32_16x16x4_f32(false, a0, false, b1, (short)0, acc01, false, false);
      acc10 = __builtin_amdgcn_wmma_f32_16x16x4_f32(false, a1, false, b0, (short)0, acc10, false, false);
      acc11 = __builtin_amdgcn_wmma_f32_16x16x4_f32(false, a1, false, b1, (short)0, acc11, false, false);
    }
  } else {
    const float* bcol0 = Bd + (n0 + l16);
    const float* bcol1 = bcol0 + 16;
    for (int k = 0; k < K; k += 4) {
      if ((k & 63) == 0) __builtin_prefetch(arow0 + k + 128, 0, 3);
      v2f a0; a0.x = arow0[k + kh];               a0.y = arow0[k + kh + 1];
      v2f a1; a1.x = arow1[k + kh];               a1.y = arow1[k + kh + 1];
      v2f b0; b0.x = bcol0[(long)(k + kh) * ldb]; b0.y = bcol0[(long)(k + kh + 1) * ldb];
      v2f b1; b1.x = bcol1[(long)(k + kh) * ldb]; b1.y = bcol1[(long)(k + kh + 1) * ldb];
      acc00 = __builtin_amdgcn_wmma_f32_16x16x4_f32(false, a0, false, b0, (short)0, acc00, false, false);
      acc01 = __builtin_amdgcn_wmma_f32_16x16x4_f32(false, a0, false, b1, (short)0, acc01, false, false);
      acc10 = __builtin_amdgcn_wmma_f32_16x16x4_f32(false, a1, false, b0, (short)0, acc10, false, false);
      acc11 = __builtin_amdgcn_wmma_f32_16x16x4_f32(false, a1, false, b1, (short)0, acc11, false, false);
    }
  }

  const int rofs = (lane >> 4) << 3;              // C layout: M = r (+8 for hi half)
  const int col0 = n0 + l16, col1 = n0 + 16 + l16;
  const float bv0 = bias ? bias[col0] : 0.f;
  const float bv1 = bias ? bias[col1] : 0.f;
#pragma unroll
  for (int r = 0; r < 8; ++r) {
    const long row0 = (long)(m0 + r + rofs) * ldc;
    const long row1 = (long)(m0 + 16 + r + rofs) * ldc;
    Cd[row0 + col0] = acc00[r] + bv0;
    Cd[row0 + col1] = acc01[r] + bv1;
    Cd[row1 + col0] = acc10[r] + bv0;
    Cd[row1 + col1] = acc11[r] + bv1;
  }
}

// ------------------------------ small utils --------------------------------
__global__ void k_fill(float* p, long n, float v) {
  for (long i = blockIdx.x * (long)blockDim.x + threadIdx.x; i < n;
       i += (long)gridDim.x * blockDim.x) p[i] = v;
}

__global__ void k_dwnorm(const float* __restrict__ dw, float* __restrict__ out) {
  if (threadIdx.x == 0 && blockIdx.x == 0) {
    float s = 0.f;
    for (int i = 0; i < Dd; ++i) s += dw[i];
    for (int i = 0; i < Dd; ++i) out[i] = dw[i] / s;
  }
}

// deg[d][dst] += w ; later dinv = (deg+1)^-1/2  (self loop weight 1)
__global__ void k_deg(const int* __restrict__ EI, const float* __restrict__ EW,
                      float* __restrict__ DEG) {
  long total = (long)Dd * Ee;
  for (long g = blockIdx.x * (long)blockDim.x + threadIdx.x; g < total;
       g += (long)gridDim.x * blockDim.x) {
    int d = (int)(g / Ee), e = (int)(g % Ee);
    int dst = EI[(long)d * 2 * Ee + Ee + e];
    atomicAdd(&DEG[d * Nn + dst], EW[(long)d * Ee + e]);
  }
}

__global__ void k_dinv(float* __restrict__ DEG) {
  long i = blockIdx.x * (long)blockDim.x + threadIdx.x;
  if (i < (long)Dd * Nn) DEG[i] = 1.f / sqrtf(DEG[i] + 1.f);
}

// G[d,n,c] = bg[d,c] + dinv[d,n]^2 * H[d,n,c]   (self-loop term + gcn bias)
__global__ void k_gcn_init(const float* __restrict__ H, float* __restrict__ G,
                           const float* __restrict__ bg,
                           const float* __restrict__ DINV, int C) {
  long total = (long)Dd * Nn * C;
  long g = blockIdx.x * (long)blockDim.x + threadIdx.x;
  if (g >= total) return;
  int d = (int)(g / ((long)Nn * C));
  long r = g % ((long)Nn * C);
  int n = (int)(r / C), c = (int)(r % C);
  float di = DINV[d * Nn + n];
  G[g] = bg[d * C + c] + di * di * H[g];
}

// G[d,dst,c] += dinv[src]*w*dinv[dst] * H[d,src,c]   (edge scatter, 4 feats/thread)
__global__ void k_scatter(const float* __restrict__ H, float* __restrict__ G,
                          const int* __restrict__ EI, const float* __restrict__ EW,
                          const float* __restrict__ DINV, int C) {
  int per = C >> 2;
  long total = (long)Dd * Ee * per;
  for (long g = blockIdx.x * (long)blockDim.x + threadIdx.x; g < total;
       g += (long)gridDim.x * blockDim.x) {
    long de = g / per;
    int cg  = (int)(g % per);
    int d   = (int)(de / Ee), e = (int)(de % Ee);
    int src = EI[(long)d * 2 * Ee + e];
    int dst = EI[(long)d * 2 * Ee + Ee + e];
    float coef = DINV[d * Nn + src] * EW[(long)d * Ee + e] * DINV[d * Nn + dst];
    const float* hs = H + ((long)d * Nn + src) * C + cg * 4;
    float*       go = G + ((long)d * Nn + dst) * C + cg * 4;
#pragma unroll
    for (int i = 0; i < 4; ++i) atomicAdd(&go[i], coef * hs[i]);
  }
}

// FUSED[n,c] = sum_d dwn[d] * GOUT[d,n,c]
__global__ void k_fuse(const float* __restrict__ GOUT, const float* __restrict__ dwn,
                       float* __restrict__ FUSED, int C) {
  long total = (long)Nn * C;
  long g = blockIdx.x * (long)blockDim.x + threadIdx.x;
  if (g >= total) return;
  float s = 0.f;
#pragma unroll
  for (int d = 0; d < Dd; ++d) s += dwn[d] * GOUT[(long)d * Nn * C + g];
  FUSED[g] = s;
}

// X[d,n,c] = relu(2*LIN + sw*FUSED[n,c]/nz[n])
__global__ void k_pre_out(const float* __restrict__ LIN, const float* __restrict__ FUSED,
                          const float* __restrict__ NZ, float* __restrict__ X,
                          int C, float sw) {
  long total = (long)Dd * Nn * C;
  long g = blockIdx.x * (long)blockDim.x + threadIdx.x;
  if (g >= total) return;
  long r = g % ((long)Nn * C);
  int n = (int)(r / C);
  float v = 2.f * LIN[g] + sw * FUSED[r] / NZ[n];
  X[g] = v > 0.f ? v : 0.f;
}

// XN[dn,:] = G[dn,:] / (||row|| + 1e-8)
__global__ __launch_bounds__(256) void k_rownorm(const float* __restrict__ G,
                                                 float* __restrict__ XN, int C) {
  long dn = blockIdx.x;
  const float* g = G + dn * C;
  float* o = XN + dn * C;
  __shared__ float red[256];
  float s = 0.f;
  for (int c = threadIdx.x; c < C; c += 256) { float v = g[c]; s += v * v; }
  red[threadIdx.x] = s; __syncthreads();
  for (int st = 128; st > 0; st >>= 1) {
    if ((int)threadIdx.x < st) red[threadIdx.x] += red[threadIdx.x + st];
    __syncthreads();
  }
  float inv = 1.f / (sqrtf(red[0]) + 1e-8f);
  for (int c = threadIdx.x; c < C; c += 256) o[c] = g[c] * inv;
}

// In-place: Lap = diag(rowsum(S)) - S   (block per (d,row))
__global__ __launch_bounds__(256) void k_laplacian(float* __restrict__ S) {
  int d = blockIdx.y, i = blockIdx.x;
  float* row = S + ((long)d * Nn + i) * Nn;
  __shared__ float red[256];
  float s = 0.f;
  for (int j = threadIdx.x; j < Nn; j += 256) s += row[j];
  red[threadIdx.x] = s; __syncthreads();
  for (int st = 128; st > 0; st >>= 1) {
    if ((int)threadIdx.x < st) red[threadIdx.x] += red[threadIdx.x + st];
    __syncthreads();
  }
  float rs = red[0];
  for (int j = threadIdx.x; j < Nn; j += 256) {
    float v = row[j];
    row[j] = (j == i) ? (rs - v) : -v;
  }
}

// Householder tridiagonalization, one 1024-thread workgroup per dim-matrix.
__global__ __launch_bounds__(1024) void k_tridiag(float* __restrict__ Abase,
                                                  float* __restrict__ dD,
                                                  float* __restrict__ dE) {
  const int n = Nn;
  int d = blockIdx.x, t = threadIdx.x;
  float* A    = Abase + (long)d * n * n;
  float* diag = dD + (long)d * n;
  float* off  = dE + (long)d * n;
  __shared__ float sv[Nn], sp[Nn], sred[Nn];
  __shared__ float s_alpha, s_beta;

  for (int k = 0; k < n - 2; ++k) {
    const int m = n - 1 - k;
    float xv = (t < m) ? A[(long)(k + 1 + t) * n + k] : 0.f;
    sv[t] = xv;
    sred[t] = xv * xv;
    __syncthreads();
    for (int s = Nn / 2; s > 0; s >>= 1) {
      if (t < s) sred[t] += sred[t + s];
      __syncthreads();
    }
    if (t == 0) {
      float sigma = sred[0], x0 = sv[0];
      float alpha = (sigma > 0.f) ? -copysignf(sqrtf(sigma), x0) : 0.f;
      float vn2 = sigma - 2.f * alpha * x0 + alpha * alpha;   // |v|^2, v0 = x0-alpha
      s_alpha = alpha;
      s_beta  = (vn2 > 1e-30f) ? 2.f / vn2 : 0.f;
      off[k]  = alpha;
      diag[k] = A[(long)k * n + k];
    }
    __syncthreads();
    float beta = s_beta;
    if (t == 0) sv[0] -= s_alpha;            // v = x - alpha*e1
    __syncthreads();

    float pv = 0.f;
    if (t < m && beta != 0.f) {
      const float* row = A + (long)(k + 1 + t) * n + (k + 1);
      float a = 0.f;
      for (int j = 0; j < m; ++j) a += row[j] * sv[j];
      pv = beta * a;
    }
    sp[t] = pv;
    sred[t] = sv[t] * pv;
    __syncthreads();
    for (int s = Nn / 2; s > 0; s >>= 1) {
      if (t < s) sred[t] += sred[t + s];
      __syncthreads();
    }
    float Kc = 0.5f * beta * sred[0];
    sp[t] = pv - Kc * sv[t];                 // w = p - K v
    __syncthreads();

    if (t < m && beta != 0.f) {
      float vi = sv[t], wi = sp[t];
      float* row = A + (long)(k + 1 + t) * n + (k + 1);
      for (int j = 0; j < m; ++j) row[j] -= vi * sp[j] + wi * sv[j];
    }
    __syncthreads();
  }
  if (t == 0) {
    diag[n - 2] = A[(long)(n - 2) * n + (n - 2)];
    off[n - 2]  = A[(long)(n - 1) * n + (n - 2)];
    diag[n - 1] = A[(long)(n - 1) * n + (n - 1)];
    off[n - 1]  = 0.f;
  }
}

// Sturm-sequence bisection: thread t finds eigenvalue #t (ascending).
__global__ __launch_bounds__(1024) void k_bisect(const float* __restrict__ dD,
                                                 const float* __restrict__ dE,
                                                 float* __restrict__ EV) {
  const int n = Nn;
  int d = blockIdx.x, t = threadIdx.x;
  __shared__ float sd[Nn], se[Nn];
  sd[t] = dD[(long)d * n + t];
  se[t] = dE[(long)d * n + t];
  __syncthreads();

  float lo = 3.4e38f, hi = -3.4e38f;
  for (int i = 0; i < n; ++i) {
    float r = (i > 0 ? fabsf(se[i - 1]) : 0.f) + (i < n - 1 ? fabsf(se[i]) : 0.f);
    lo = fminf(lo, sd[i] - r);
    hi = fmaxf(hi, sd[i] + r);
  }
  for (int it = 0; it < 64; ++it) {
    float mid = 0.5f * (lo + hi);
    int cnt = 0;
    float q = 1.f;
    for (int i = 0; i < n; ++i) {
      float e2 = (i > 0) ? se[i - 1] * se[i - 1] : 0.f;
      q = sd[i] - mid - e2 / q;
      if (q < 0.f) ++cnt;
      if (fabsf(q) < 1e-30f) q = -1e-30f;
    }
    if (cnt <= t) lo = mid; else hi = mid;
  }
  EV[(long)d * n + t] = 0.5f * (lo + hi);
}

// eigvalsh output is ascending -> argsort = identity -> flags = cumsum(diff>thr)
__global__ void k_flags(const float* __restrict__ EV, int* __restrict__ FL, float thr) {
  int d = blockIdx.x;
  if (threadIdx.x == 0) {
    int c = 0;
    FL[(long)d * Nn] = 0;
    for (int i = 1; i < Nn; ++i) {
      if (EV[(long)d * Nn + i] - EV[(long)d * Nn + i - 1] > thr) ++c;
      FL[(long)d * Nn + i] = c;
    }
  }
}

__global__ void k_classsum(const float* __restrict__ G, const int* __restrict__ FL,
                           float* __restrict__ CS, float* __restrict__ CC, int C) {
  long total = (long)Dd * Nn * C;
  long g = blockIdx.x * (long)blockDim.x + threadIdx.x;
  if (g >= total) return;
  int d = (int)(g / ((long)Nn * C));
  long r = g % ((long)Nn * C);
  int n = (int)(r / C), c = (int)(r % C);
  int f = FL[d * Nn + n];
  atomicAdd(&CS[((long)d * Nn + f) * C + c], G[g]);
  if (c == 0) atomicAdd(&CC[d * Nn + f], 1.f);
}

__global__ void k_gatheravg(const int* __restrict__ FL, const float* __restrict__ CS,
                            const float* __restrict__ CC, float* __restrict__ AVG, int C) {
  long total = (long)Dd * Nn * C;
  long g = blockIdx.x * (long)blockDim.x + threadIdx.x;
  if (g >= total) return;
  int d = (int)(g / ((long)Nn * C));
  long r = g % ((long)Nn * C);
  int n = (int)(r / C), c = (int)(r % C);
  int f = FL[d * Nn + n];
  AVG[g] = CS[((long)d * Nn + f) * C + c] / fmaxf(CC[d * Nn + f], 1.f);
}

// FIR = relu(2*FIR + k_save*LB)   (in-place)
__global__ void k_behout(float* __restrict__ FIR, const float* __restrict__ LB,
                         const float* __restrict__ ks, long total) {
  long g = blockIdx.x * (long)blockDim.x + threadIdx.x;
  if (g >= total) return;
  float v = 2.f * FIR[g] + ks[0] * LB[g];
  FIR[g] = v > 0.f ? v : 0.f;
}

__global__ void k_final(const float* __restrict__ FIR, float* __restrict__ out) {
  long total = (long)Nn * Oo;
  long g = blockIdx.x * (long)blockDim.x + threadIdx.x;
  if (g >= total) return;
  float s = 1e-8f;
#pragma unroll
  for (int d = 0; d < Dd; ++d) s += FIR[(long)d * Nn * Oo + g];
  out[g] = s;
}

// ------------------------------ host driver --------------------------------
extern "C" void kernel_launch(void* const* d_in, const int* in_sizes, int n_in,
                              void* d_out, int out_size, void* d_ws, size_t ws_size,
                              hipStream_t stream) {
  (void)in_sizes; (void)n_in; (void)out_size; (void)ws_size;

  const float* dims[Dd] = {(const float*)d_in[0], (const float*)d_in[1],
                           (const float*)d_in[2], (const float*)d_in[3],
                           (const float*)d_in[4]};
  const int*   EI  = (const int*)d_in[5];
  const float* EW  = (const float*)d_in[6];
  const float* NZ  = (const float*)d_in[7];
  const float* GW1 = (const float*)d_in[8];  const float* GB1 = (const float*)d_in[9];
  const float* GW2 = (const float*)d_in[10]; const float* GB2 = (const float*)d_in[11];
  const float* GW3 = (const float*)d_in[12]; const float* GB3 = (const float*)d_in[13];
  const float* GWB = (const float*)d_in[14]; const float* GBB = (const float*)d_in[15];
  const float* WP[3] = {(const float*)d_in[16], (const float*)d_in[18], (const float*)d_in[20]};
  const float* BP[3] = {(const float*)d_in[17], (const float*)d_in[19], (const float*)d_in[21]};
  const float* WB[3] = {(const float*)d_in[22], (const float*)d_in[24], (const float*)d_in[26]};
  const float* BB[3] = {(const float*)d_in[23], (const float*)d_in[25], (const float*)d_in[27]};
  const float* DW  = (const float*)d_in[28];
  const float* KS  = (const float*)d_in[29];

  // ---- bump-allocate scratch ----
  char* base = (char*)d_ws;
  size_t off = 0;
  auto carve = [&](long nfloats) -> float* {
    float* p = (float*)(base + off);
    off += ((size_t)nfloats * 4 + 255) & ~(size_t)255;
    return p;
  };
  float* XBUF  = carve((long)Dd * Nn * INF);   // layer inputs / pre outputs
  float* LIN   = carve((long)Dd * Nn * Hh);
  float* HB    = carve((long)Dd * Nn * Hh);
  float* GOUT  = carve((long)Dd * Nn * Hh);
  float* FUSED = carve((long)Nn * Hh);
  float* DINV  = carve((long)Dd * Nn);
  float* DWN   = carve(8);
  float* GBA   = carve((long)Dd * Nn * Oo);    // beh g ping
  float* GBB2  = carve((long)Dd * Nn * Oo);    // beh g pong
  float* FIR   = carve((long)Dd * Nn * Oo);
  float* XN    = carve((long)Dd * Nn * Oo);
  float* SB    = carve((long)Dd * Nn * Nn);    // similarity / Laplacian (in place)
  float* TD    = carve((long)Dd * Nn);
  float* TE    = carve((long)Dd * Nn);
  float* EV    = carve((long)Dd * Nn);
  int*   FLG   = (int*)carve((long)Dd * Nn);
  float* CSUM  = carve((long)Dd * Nn * Oo);
  float* CCNT  = carve((long)Dd * Nn);
  float* AVG   = carve((long)Dd * Nn * Oo);
  float* LB    = carve((long)Dd * Nn * Oo);

  auto gemm = [&](const float* A, int lda, long sA, const float* B, int ldb, long sB,
                  float* C, int ldc, long sC, const float* bias,
                  int M, int Ncols, int K, int transB) {
    dim3 grid(Ncols / 64, M / 64, Dd);
    k_gemm<<<grid, 128, 0, stream>>>(A, lda, sA, B, ldb, sB, C, ldc, sC, bias, K, transB);
  };

  // ---- stack dims into [D,N,IN] ----
  for (int d = 0; d < Dd; ++d)
    hipMemcpyAsync(XBUF + (long)d * Nn * INF, dims[d], (size_t)Nn * INF * 4,
                   hipMemcpyDeviceToDevice, stream);

  k_dwnorm<<<1, 32, 0, stream>>>(DW, DWN);

  // ---- degrees -> dinv (edges identical across all layers) ----
  k_fill<<<cdiv((long)Dd * Nn, 256), 256, 0, stream>>>(DINV, (long)Dd * Nn, 0.f);
  k_deg<<<2048, 256, 0, stream>>>(EI, EW, DINV);
  k_dinv<<<cdiv((long)Dd * Nn, 256), 256, 0, stream>>>(DINV);

  // ---- 3 pre layers ----
  const float saveW[3] = {1.0f, 0.36787944117144233f, 0.1353352832366127f};
  const int   inC[3]   = {INF, Hh, Oo};
  const int   outC[3]  = {Hh, Oo, Oo};
  const float* GWs[3]  = {GW1, GW2, GW3};
  const float* GBs[3]  = {GB1, GB2, GB3};

  for (int l = 0; l < 3; ++l) {
    const int ci = inC[l], co = outC[l];
    // lin = x @ Wl + bl   (shared Wl -> strideB 0)
    gemm(XBUF, ci, (long)Nn * ci, WP[l], co, 0, LIN, co, (long)Nn * co, BP[l],
         Nn, co, ci, 0);
    // h = lin @ Wg[d]
    gemm(LIN, co, (long)Nn * co, GWs[l], co, (long)co * co, HB, co, (long)Nn * co,
         nullptr, Nn, co, co, 0);
    // GCN: self loop + bias, then edge scatter
    k_gcn_init<<<cdiv((long)Dd * Nn * co, 256), 256, 0, stream>>>(HB, GOUT, GBs[l], DINV, co);
    k_scatter<<<8192, 256, 0, stream>>>(HB, GOUT, EI, EW, DINV, co);
    // fused dim mix + relu
    k_fuse<<<cdiv((long)Nn * co, 256), 256, 0, stream>>>(GOUT, DWN, FUSED, co);
    k_pre_out<<<cdiv((long)Dd * Nn * co, 256), 256, 0, stream>>>(LIN, FUSED, NZ, XBUF,
                                                                 co, saveW[l]);
  }

  // ---- beh stage: FIR = L3, gin = L3 ----
  hipMemcpyAsync(FIR, XBUF, (size_t)Dd * Nn * Oo * 4, hipMemcpyDeviceToDevice, stream);

  float* gin = XBUF;
  float* gping[2] = {GBA, GBB2};
  for (int l = 0; l < 3; ++l) {
    float* g = gping[l & 1];
    // g = gcn(gin, gcn_wb, gcn_bb)
    gemm(gin, Oo, (long)Nn * Oo, GWB, Oo, (long)Oo * Oo, HB, Oo, (long)Nn * Oo,
         nullptr, Nn, Oo, Oo, 0);
    k_gcn_init<<<cdiv((long)Dd * Nn * Oo, 256), 256, 0, stream>>>(HB, g, GBB, DINV, Oo);
    k_scatter<<<8192, 256, 0, stream>>>(HB, g, EI, EW, DINV, Oo);

    // spectral flags: Xn, S = Xn @ Xn^T, Laplacian, eigvals, eigengap classes
    k_rownorm<<<Dd * Nn, 256, 0, stream>>>(g, XN, Oo);
    gemm(XN, Oo, (long)Nn * Oo, XN, Oo, (long)Nn * Oo, SB, Nn, (long)Nn * Nn,
         nullptr, Nn, Nn, Oo, 1);
    k_laplacian<<<dim3(Nn, Dd), 256, 0, stream>>>(SB);
    k_tridiag<<<Dd, 1024, 0, stream>>>(SB, TD, TE);
    k_bisect<<<Dd, 1024, 0, stream>>>(TD, TE, EV);
    k_flags<<<Dd, 32, 0, stream>>>(EV, FLG, 0.5f);

    // class means gathered back per node
    k_fill<<<cdiv((long)Dd * Nn * Oo, 256), 256, 0, stream>>>(CSUM, (long)Dd * Nn * Oo, 0.f);
    k_fill<<<cdiv((long)Dd * Nn, 256), 256, 0, stream>>>(CCNT, (long)Dd * Nn, 0.f);
    k_classsum<<<cdiv((long)Dd * Nn * Oo, 256), 256, 0, stream>>>(g, FLG, CSUM, CCNT, Oo);
    k_gatheravg<<<cdiv((long)Dd * Nn * Oo, 256), 256, 0, stream>>>(FLG, CSUM, CCNT, AVG, Oo);

    // lin = avg @ Wl + bl ; FIR = relu(2*FIR + k_save*lin)
    gemm(AVG, Oo, (long)Nn * Oo, WB[l], Oo, 0, LB, Oo, (long)Nn * Oo, BB[l],
         Nn, Oo, Oo, 0);
    k_behout<<<cdiv((long)Dd * Nn * Oo, 256), 256, 0, stream>>>(FIR, LB, KS,
                                                                (long)Dd * Nn * Oo);
    gin = g;
  }

  k_final<<<cdiv((long)Nn * Oo, 256), 256, 0, stream>>>(FIR, (float*)d_out);
}